// EntMaxBisectDelegator_57243324121572
// MI455X (gfx1250) — compile-verified
//
#include <hip/hip_runtime.h>
#include <cstdint>
#include <cstddef>

// Row length is fixed by the reference: (4, 4096, 2048), entmax over dim=-1.
#define D        2048
#define WPB      8                // waves per block (wave32 => 256 threads)
#define ELPL     (D / 32)         // 64 f32 elements per lane
#define CH       (ELPL / 4)       // 16 float4 chunks per lane
#define NIT      50               // bisection iterations (entmax default)
// (1/d)^(alpha-1) = sqrt(1/2048)
#define GP_INVD  0.022097086912079608f

#if defined(__HIP_DEVICE_COMPILE__) && __has_builtin(__builtin_amdgcn_global_load_async_to_lds_b128)
#define HAVE_ASYNC 1
#else
#define HAVE_ASYNC 0
#endif

#if defined(__HIP_DEVICE_COMPILE__) && __has_builtin(__builtin_amdgcn_update_dpp)
#define HAVE_DPP 1
#else
#define HAVE_DPP 0
#endif

#if HAVE_ASYNC
typedef int v4i __attribute__((vector_size(16)));
typedef __attribute__((address_space(1))) v4i gv4i;   // global int4
typedef __attribute__((address_space(3))) v4i lv4i;   // LDS int4
#endif

__device__ __forceinline__ void wait_async_0() {
#if defined(__HIP_DEVICE_COMPILE__)
#if __has_builtin(__builtin_amdgcn_s_wait_asynccnt)
  __builtin_amdgcn_s_wait_asynccnt(0);
#else
  asm volatile("s_wait_asynccnt 0" ::: "memory");
#endif
#endif
}

__device__ __forceinline__ void wait_ds_0() {
#if defined(__HIP_DEVICE_COMPILE__)
#if __has_builtin(__builtin_amdgcn_s_wait_dscnt)
  __builtin_amdgcn_s_wait_dscnt(0);
#else
  asm volatile("s_wait_dscnt 0" ::: "memory");
#endif
#endif
}

#if HAVE_DPP
// DPP lane permute (full-wave involutions, EXEC all ones => exact).
// ctrl: 0xB1 = quad_perm xor1, 0x4E = quad_perm xor2,
//       0x141 = row_half_mirror (xor within 8), 0x140 = row_mirror (xor within 16)
template <int CTRL>
__device__ __forceinline__ float dpp_xchg(float v) {
  return __int_as_float(__builtin_amdgcn_update_dpp(
      __float_as_int(v), __float_as_int(v), CTRL, 0xF, 0xF, false));
}
#endif

__device__ __forceinline__ float wave_reduce_add(float v) {
#if HAVE_DPP
  v += dpp_xchg<0xB1>(v);          // xor 1
  v += dpp_xchg<0x4E>(v);          // xor 2
  v += dpp_xchg<0x141>(v);         // xor within 8
  v += dpp_xchg<0x140>(v);         // xor within 16
  v += __shfl_xor(v, 16, 32);      // cross 16-lane halves
  return v;
#else
  #pragma unroll
  for (int off = 16; off > 0; off >>= 1) v += __shfl_xor(v, off, 32);
  return v;
#endif
}

__device__ __forceinline__ float wave_reduce_max(float v) {
#if HAVE_DPP
  v = fmaxf(v, dpp_xchg<0xB1>(v));
  v = fmaxf(v, dpp_xchg<0x4E>(v));
  v = fmaxf(v, dpp_xchg<0x141>(v));
  v = fmaxf(v, dpp_xchg<0x140>(v));
  v = fmaxf(v, __shfl_xor(v, 16, 32));
  return v;
#else
  #pragma unroll
  for (int off = 16; off > 0; off >>= 1) v = fmaxf(v, __shfl_xor(v, off, 32));
  return v;
#endif
}

__global__ __launch_bounds__(WPB * 32)
void entmax_bisect_kernel(const float* __restrict__ X,
                          float* __restrict__ Out,
                          int nrows) {
  // One private 8KB row buffer per wave: 8 waves * 2048 f32 = 64 KB LDS/block.
  __shared__ float4 lbuf[WPB * (D / 4)];

  const int lane   = threadIdx.x & 31;
  const int wave   = threadIdx.x >> 5;
  const int nwaves = gridDim.x * WPB;
  float4* mybuf    = &lbuf[wave * (D / 4)];

  // Async DMA of one full row (8KB) into this wave's LDS buffer.
  auto prefetch = [&](int r) {
#if HAVE_ASYNC
    const float4* src = (const float4*)(X + (size_t)r * D);
    #pragma unroll
    for (int c = 0; c < CH; ++c) {
      const int idx = c * 32 + lane;          // float4 index within the row
      __builtin_amdgcn_global_load_async_to_lds_b128(
          (gv4i*)(src + idx),
          (lv4i*)(mybuf + idx),
          0, 0);
    }
#else
    (void)r;
#endif
  };

  int row = blockIdx.x * WPB + wave;
  if (row < nrows) prefetch(row);

  for (; row < nrows; row += nwaves) {
    float xs[ELPL];

#if HAVE_ASYNC
    wait_async_0();                            // row resident in LDS
    #pragma unroll
    for (int c = 0; c < CH; ++c) {
      float4 v = mybuf[c * 32 + lane];         // ds_load_b128
      xs[4*c+0] = v.x * 0.5f; xs[4*c+1] = v.y * 0.5f;
      xs[4*c+2] = v.z * 0.5f; xs[4*c+3] = v.w * 0.5f;
    }
    wait_ds_0();                               // LDS reads landed in VGPRs
    asm volatile("" ::: "memory");
    if (row + nwaves < nrows) prefetch(row + nwaves);  // overlap with bisection
#else
    const float4* src = (const float4*)(X + (size_t)row * D);
    #pragma unroll
    for (int c = 0; c < CH; ++c) {
      float4 v = src[c * 32 + lane];
      xs[4*c+0] = v.x * 0.5f; xs[4*c+1] = v.y * 0.5f;
      xs[4*c+2] = v.z * 0.5f; xs[4*c+3] = v.w * 0.5f;
    }
#endif

    // ---- row max ----
    float m = xs[0];
    #pragma unroll
    for (int i = 1; i < ELPL; ++i) m = fmaxf(m, xs[i]);
    m = wave_reduce_max(m);

    float tau_lo = m - 1.0f;                   // _gp(1, alpha)
    const float tau_hi = m - GP_INVD;          // _gp(1/d, alpha)

    // ---- f_lo = sum(relu(Xs - tau_lo)^2) - 1 ----
    {
      float s0 = 0.f, s1 = 0.f, s2 = 0.f, s3 = 0.f;
      #pragma unroll
      for (int i = 0; i < ELPL; i += 4) {
        float r0 = fmaxf(xs[i+0] - tau_lo, 0.f);
        float r1 = fmaxf(xs[i+1] - tau_lo, 0.f);
        float r2 = fmaxf(xs[i+2] - tau_lo, 0.f);
        float r3 = fmaxf(xs[i+3] - tau_lo, 0.f);
        s0 = fmaf(r0, r0, s0); s1 = fmaf(r1, r1, s1);
        s2 = fmaf(r2, r2, s2); s3 = fmaf(r3, r3, s3);
      }
      m = wave_reduce_add((s0 + s1) + (s2 + s3));
    }
    const float f_lo = m - 1.0f;

    // ---- 50 bisection steps; alpha=1.5 => exponent is exactly 2 ----
    float dm    = tau_hi - tau_lo;
    float tau_m = tau_lo;
    for (int it = 0; it < NIT; ++it) {
      dm *= 0.5f;
      tau_m = tau_lo + dm;
      float a0 = 0.f, a1 = 0.f, a2 = 0.f, a3 = 0.f;
      #pragma unroll
      for (int i = 0; i < ELPL; i += 4) {
        float r0 = fmaxf(xs[i+0] - tau_m, 0.f);
        float r1 = fmaxf(xs[i+1] - tau_m, 0.f);
        float r2 = fmaxf(xs[i+2] - tau_m, 0.f);
        float r3 = fmaxf(xs[i+3] - tau_m, 0.f);
        a0 = fmaf(r0, r0, a0); a1 = fmaf(r1, r1, a1);
        a2 = fmaf(r2, r2, a2); a3 = fmaf(r3, r3, a3);
      }
      float fm = wave_reduce_add((a0 + a1) + (a2 + a3)) - 1.0f;
      // reference keeps f_lo fixed and moves tau_lo when signs agree
      tau_lo = (fm * f_lo >= 0.0f) ? tau_m : tau_lo;
    }

    // ---- p = relu(Xs - tau_m)^2, then renormalize (ensure_sum_one) ----
    float t0 = 0.f, t1 = 0.f, t2 = 0.f, t3 = 0.f;
    #pragma unroll
    for (int i = 0; i < ELPL; i += 4) {
      float r0 = fmaxf(xs[i+0] - tau_m, 0.f);
      float r1 = fmaxf(xs[i+1] - tau_m, 0.f);
      float r2 = fmaxf(xs[i+2] - tau_m, 0.f);
      float r3 = fmaxf(xs[i+3] - tau_m, 0.f);
      xs[i+0] = r0 * r0; xs[i+1] = r1 * r1;
      xs[i+2] = r2 * r2; xs[i+3] = r3 * r3;
      t0 = fmaf(r0, r0, t0); t1 = fmaf(r1, r1, t1);
      t2 = fmaf(r2, r2, t2); t3 = fmaf(r3, r3, t3);
    }
    const float S    = wave_reduce_add((t0 + t1) + (t2 + t3));
    const float invS = 1.0f / S;

    float4* orow = (float4*)(Out + (size_t)row * D);
    #pragma unroll
    for (int c = 0; c < CH; ++c) {
      float4 v;
      v.x = xs[4*c+0] * invS; v.y = xs[4*c+1] * invS;
      v.z = xs[4*c+2] * invS; v.w = xs[4*c+3] * invS;
      orow[c * 32 + lane] = v;
    }
  }
}

extern "C" void kernel_launch(void* const* d_in, const int* in_sizes, int n_in,
                              void* d_out, int out_size, void* d_ws, size_t ws_size,
                              hipStream_t stream) {
  (void)n_in; (void)out_size; (void)d_ws; (void)ws_size;
  const float* X = (const float*)d_in[0];
  float* Out     = (float*)d_out;

  const int nrows = in_sizes[0] / D;           // 16384 for the reference shape
  // ~2 rows per wave so steady-state row loads hide under the bisection loop.
  int waves  = (nrows + 1) / 2;
  int blocks = (waves + WPB - 1) / WPB;
  if (blocks < 1) blocks = 1;

  entmax_bisect_kernel<<<blocks, WPB * 32, 0, stream>>>(X, Out, nrows);
}